// MoEKANConvBase_90520730730681
// MI455X (gfx1250) — compile-verified
//
#include <hip/hip_runtime.h>
#include <hip/hip_bf16.h>
#include <math.h>

typedef __attribute__((ext_vector_type(2))) float v2f;
typedef __attribute__((ext_vector_type(8))) float v8f;

#define B_    64
#define CIN   32
#define COUT  32
#define HH    64
#define WW    64
#define HW    (HH*WW)
#define NE    16
#define TOPK  4

// ------------------------------------------------------------------
// Kernel A: gate_x[b][c] = mean over H*W of x[b][c][:][:]
// ------------------------------------------------------------------
__global__ __launch_bounds__(256) void gatex_kernel(const float* __restrict__ x,
                                                    float* __restrict__ gate_x) {
    int bc = blockIdx.x;
    const float* p = x + (size_t)bc * HW;
    float s = 0.f;
    for (int i = threadIdx.x; i < HW; i += 256) s += p[i];
    __shared__ float red[256];
    red[threadIdx.x] = s;
    __syncthreads();
    for (int w = 128; w > 0; w >>= 1) {
        if (threadIdx.x < w) red[threadIdx.x] += red[threadIdx.x + w];
        __syncthreads();
    }
    if (threadIdx.x == 0) gate_x[bc] = red[0] * (1.0f / HW);
}

// ------------------------------------------------------------------
// Kernel B: full gating pipeline, one block of 64 threads (1 row each)
// ------------------------------------------------------------------
__device__ inline float cv_squared16(const float* v) {
    float m = 0.f;
    for (int i = 0; i < NE; i++) m += v[i];
    m *= (1.0f / NE);
    float var = 0.f;
    for (int i = 0; i < NE; i++) { float d = v[i] - m; var += d * d; }
    var *= (1.0f / (NE - 1));          // ddof = 1
    return var / (m * m + 1e-10f);
}

__global__ __launch_bounds__(64) void gating_kernel(const float* __restrict__ gate_x,
                                                    const float* __restrict__ noise,
                                                    const float* __restrict__ w_gate,
                                                    const float* __restrict__ w_noise,
                                                    float* __restrict__ gates_out,
                                                    float* __restrict__ loss_out) {
    int b = threadIdx.x;               // 0..63
    float clean[NE], stdv[NE], noisy[NE], logit[NE], gat[NE], prob[NE];

    for (int e = 0; e < NE; e++) {
        float c = 0.f, n = 0.f;
        for (int ci = 0; ci < CIN; ci++) {
            float g = gate_x[b * CIN + ci];
            c += g * w_gate[ci * NE + e];
            n += g * w_noise[ci * NE + e];
        }
        clean[e] = c;
        float sp = (n > 20.f) ? n : log1pf(expf(n));   // softplus
        stdv[e] = sp + 0.01f;
        noisy[e] = c + noise[b * NE + e] * stdv[e];
    }
    // softmax
    float mx = noisy[0];
    for (int e = 1; e < NE; e++) mx = fmaxf(mx, noisy[e]);
    float sum = 0.f;
    for (int e = 0; e < NE; e++) { logit[e] = expf(noisy[e] - mx); sum += logit[e]; }
    float inv = 1.0f / sum;
    for (int e = 0; e < NE; e++) logit[e] *= inv;

    // top-(k+1) selection
    float tmp[NE];
    for (int e = 0; e < NE; e++) tmp[e] = logit[e];
    float tv[TOPK + 1];
    int   ti[TOPK + 1];
    for (int j = 0; j < TOPK + 1; j++) {
        int bi = 0; float bv = tmp[0];
        for (int e = 1; e < NE; e++) { if (tmp[e] > bv) { bv = tmp[e]; bi = e; } }
        tv[j] = bv; ti[j] = bi; tmp[bi] = -1e30f;
    }
    float ssum = tv[0] + tv[1] + tv[2] + tv[3] + 1e-6f;
    for (int e = 0; e < NE; e++) gat[e] = 0.f;
    for (int j = 0; j < TOPK; j++) gat[ti[j]] = tv[j] / ssum;
    float thr_in = tv[TOPK];           // (k+1)-th logit
    float thr_out = tv[TOPK - 1];      // k-th logit

    const float inv_sqrt2 = 0.70710678118654752f;
    for (int e = 0; e < NE; e++) {
        float thr = (noisy[e] > thr_in) ? thr_in : thr_out;
        float z = (clean[e] - thr) / stdv[e];
        prob[e] = 0.5f * (1.0f + erff(z * inv_sqrt2));
    }

    __shared__ float sProb[B_][NE];
    __shared__ float sGate[B_][NE];
    for (int e = 0; e < NE; e++) {
        sProb[b][e] = prob[e];
        sGate[b][e] = gat[e];
        gates_out[b * NE + e] = gat[e];
    }
    __syncthreads();
    __shared__ float sLoad[NE], sImp[NE];
    if (b < NE) {
        float l = 0.f, im = 0.f;
        for (int r = 0; r < B_; r++) { l += sProb[r][b]; im += sGate[r][b]; }
        sLoad[b] = l; sImp[b] = im;
    }
    __syncthreads();
    if (b == 0) {
        float loss = (cv_squared16(sImp) + cv_squared16(sLoad)) * 0.01f;
        loss_out[0] = loss;
    }
}

// ------------------------------------------------------------------
// Kernel C: per-sample combined-weight conv3x3 via V_WMMA_F32_16X16X4_F32.
// grid = B*8 blocks (8 rows each), 256 threads = 8 wave32.
//  * x tile transposed to [r][col][cin] (cin innermost, stride 34 to kill
//    bank conflicts), zero halo -> every WMMA fragment is one aligned
//    ds_load_b64, no bounds checks, no register shuffles.
//  * tile filled with GLOBAL_LOAD_ASYNC_TO_LDS_B32 (per-lane scattered LDS
//    dest, contiguous global side), overlapped with the weight combine.
// ------------------------------------------------------------------
#define CSTRIDE 34                       // padded cin stride (dwords)
#define NCOL    72                       // cols: [3]=left halo, [4..67]=data, [68]=right halo
#define XPLANE  (NCOL * CSTRIDE)         // floats per r-plane

__global__ __launch_bounds__(256) void moeconv_kernel(const float* __restrict__ x,
                                                      const float* __restrict__ expert_w,
                                                      const float* __restrict__ expert_b,
                                                      const float* __restrict__ gates,
                                                      float* __restrict__ y) {
    __shared__ float xs[10 * XPLANE];            // [r][col][cin pad 34] : ~95.6 KB
    __shared__ float wc[9 * COUT * CSTRIDE];     // [tap][cout][cin pad 34] : ~38.3 KB
    __shared__ float sBias[COUT];
    __shared__ float sG[NE];

    int b   = blockIdx.x >> 3;
    int rg  = blockIdx.x & 7;
    int tid = threadIdx.x;

    if (tid < NE) sG[tid] = gates[b * NE + tid];

    // zero halo columns (sw == -1 and sw == 64) for every (r, cin)
    for (int i = tid; i < 10 * CIN; i += 256) {
        int r = i >> 5, cin = i & 31;
        xs[r * XPLANE + 3  * CSTRIDE + cin] = 0.f;
        xs[r * XPLANE + 68 * CSTRIDE + cin] = 0.f;
    }
    // zero out-of-image row planes (block-uniform branches)
    if (rg == 0) for (int i = tid; i < XPLANE; i += 256) xs[i] = 0.f;
    if (rg == 7) for (int i = tid; i < XPLANE; i += 256) xs[9 * XPLANE + i] = 0.f;

    const float* xb = x + (size_t)b * CIN * HW;

    // async-stage valid x rows into LDS, transposing [cin][w] -> [w][cin].
    // lanes are consecutive in w  -> global side is 128B-contiguous per wave;
    // LDS side is per-lane scattered (stride CSTRIDE dwords).
    {
        int roff = (rg == 0) ? 1 : 0;
        int rend = (rg == 7) ? 9 : 10;
        int w  = tid & 63;               // fixed per thread
        int cg = tid >> 6;               // cin slot 0..3
        for (int r = roff; r < rend; ++r) {
            int sh = rg * 8 - 1 + r;
            unsigned ldsbase = (unsigned)(uintptr_t)&xs[r * XPLANE + (w + 4) * CSTRIDE];
            const float* gbase = xb + (size_t)sh * WW + w;
#pragma unroll
            for (int j = 0; j < 8; ++j) {
                int cin = cg + j * 4;
                unsigned ldsa = ldsbase + (unsigned)(cin * 4);
                const float* ga = gbase + (size_t)cin * HW;
                asm volatile("global_load_async_to_lds_b32 %0, %1, off"
                             :: "v"(ldsa), "v"(ga) : "memory");
            }
        }
    }
    __syncthreads();    // sG + zero-fills visible (async DMA still in flight)

    if (tid < COUT) {
        float bb = 0.f;
        for (int e = 0; e < NE; e++) bb += sG[e] * expert_b[e * COUT + tid];
        sBias[tid] = bb;
    }
    // combine expert weights into LDS as [tap][cout][cin] (overlaps the DMA)
    for (int idx = tid; idx < 9 * COUT * CIN; idx += 256) {
        int cin  = idx & 31;
        int cout = (idx >> 5) & 31;
        int tap  = idx >> 10;
        float a = 0.f;
        for (int e = 0; e < NE; e++)
            a += sG[e] * expert_w[((size_t)(e * COUT + cout) * CIN + cin) * 9 + tap];
        wc[(tap * COUT + cout) * CSTRIDE + cin] = a;
    }

    asm volatile("s_wait_asynccnt 0" ::: "memory");  // my waves' DMA done
    __syncthreads();                                 // everyone's DMA + wc done

    int wave = tid >> 5;
    int lane = tid & 31;
    int sel  = lane >> 4;      // K half: 0 -> K={0,1}, 1 -> K={2,3}
    int ln   = lane & 15;
    int coutBase = (wave & 1) * 16;
    int wBase    = (wave >> 1) * 16;
    int wpix     = wBase + ln;

    float* yb = y + (size_t)b * COUT * HW;

    for (int row = 0; row < 8; row++) {
        int h = rg * 8 + row;
        v8f acc = {0.f, 0.f, 0.f, 0.f, 0.f, 0.f, 0.f, 0.f};

#pragma unroll
        for (int tap = 0; tap < 9; tap++) {
            int dy = tap / 3 - 1;
            int dx = tap % 3 - 1;
            // both fragments: lane's two K-values are adjacent -> ds_load_b64
            const float* xrow = &xs[(row + 1 + dy) * XPLANE
                                    + (4 + wpix + dx) * CSTRIDE + sel * 2];
            const float* wrow = &wc[(tap * COUT + coutBase + ln) * CSTRIDE + sel * 2];
#pragma unroll
            for (int c4 = 0; c4 < 8; c4++) {
                v2f bv = *(const v2f*)(xrow + c4 * 4);
                v2f av = *(const v2f*)(wrow + c4 * 4);
                acc = __builtin_amdgcn_wmma_f32_16x16x4_f32(
                        false, av, false, bv, (short)0, acc, false, false);
            }
        }
        // C/D layout: VGPR r -> M = r + 8*sel (cout), N = ln (pixel)
#pragma unroll
        for (int r = 0; r < 8; r++) {
            int cout = coutBase + r + 8 * sel;
            yb[(size_t)cout * HW + h * WW + wBase + ln] = acc[r] + sBias[cout];
        }
    }
}

// ------------------------------------------------------------------
extern "C" void kernel_launch(void* const* d_in, const int* in_sizes, int n_in,
                              void* d_out, int out_size, void* d_ws, size_t ws_size,
                              hipStream_t stream) {
    const float* x        = (const float*)d_in[0];
    const float* noise    = (const float*)d_in[1];
    const float* w_gate   = (const float*)d_in[2];
    const float* w_noise  = (const float*)d_in[3];
    const float* expert_w = (const float*)d_in[4];
    const float* expert_b = (const float*)d_in[5];

    float* y    = (float*)d_out;                       // (B, COUT, H, W) flat
    float* loss = y + (size_t)B_ * COUT * HW;          // last element

    float* ws     = (float*)d_ws;
    float* gate_x = ws;                                // B*CIN  = 2048 floats
    float* gatesW = ws + B_ * CIN;                     // B*NE   = 1024 floats

    gatex_kernel<<<B_ * CIN, 256, 0, stream>>>(x, gate_x);
    gating_kernel<<<1, 64, 0, stream>>>(gate_x, noise, w_gate, w_noise, gatesW, loss);
    moeconv_kernel<<<B_ * 8, 256, 0, stream>>>(x, expert_w, expert_b, gatesW, y);
}